// EdgeGATLayer_75952201662544
// MI455X (gfx1250) — compile-verified
//
#include <hip/hip_runtime.h>
#include <hip/hip_bf16.h>

#define N_NODES 50000
#define N_EDGES 800000
#define DIN     256
#define EDIM    16
#define HC      128   // H*C
#define CH      64    // C per head
#define NEG_SLOPE 0.2f

typedef float v2f __attribute__((ext_vector_type(2)));
typedef float v8f __attribute__((ext_vector_type(8)));

// ---------------------------------------------------------------------------
// K1: fold the two edge linear layers:  W_comb = W_ep @ W_e  [16,128],
//     b_comb = b_ep @ W_e [128].  Tiny (1M FLOP), one block.
// ---------------------------------------------------------------------------
__global__ void combine_weights(const float* __restrict__ W_ep,
                                const float* __restrict__ b_ep,
                                const float* __restrict__ W_e,
                                float* __restrict__ W_comb,
                                float* __restrict__ b_comb) {
  int n = blockIdx.x * blockDim.x + threadIdx.x;
  if (n >= HC) return;
  for (int k = 0; k < EDIM; ++k) {
    float s = 0.f;
    for (int d = 0; d < DIN; ++d) s += W_ep[k * DIN + d] * W_e[d * HC + n];
    W_comb[k * HC + n] = s;
  }
  float sb = 0.f;
  for (int d = 0; d < DIN; ++d) sb += b_ep[d] * W_e[d * HC + n];
  b_comb[n] = sb;
}

// ---------------------------------------------------------------------------
// K2/K3: Y[M,128] = X[M,256] @ W[256,128] + bias, fp32 WMMA 16x16x4.
// One wave = one 16x16 output tile; threadIdx.y (0..7) = column tile.
// A layout (16x4 f32): lane m = lane&15, k = 2*(lane>>4)+j.
// B layout (4x16 f32): lane n = lane&15, k = 2*(lane>>4)+j.
// C/D (16x16 f32):     row M = j + 8*(lane>>4), col N = lane&15.
// ---------------------------------------------------------------------------
__global__ __launch_bounds__(256)
void gemm_bias_wmma(const float* __restrict__ X, const float* __restrict__ W,
                    const float* __restrict__ bias, float* __restrict__ Y,
                    int K) {
  const int lane   = threadIdx.x;       // 0..31
  const int half   = lane >> 4;
  const int l15    = lane & 15;
  const int kgrp   = half << 1;
  const int rowBase = blockIdx.x << 4;
  const int n       = (threadIdx.y << 4) + l15;
  const int m       = rowBase + l15;

  v8f acc = {};
  for (int k0 = 0; k0 < K; k0 += 4) {
    v2f a = *(const v2f*)(X + m * K + k0 + kgrp);      // k, k+1 contiguous
    v2f b;
    b.x = W[(k0 + kgrp)     * HC + n];
    b.y = W[(k0 + kgrp + 1) * HC + n];
    acc = __builtin_amdgcn_wmma_f32_16x16x4_f32(
        /*neg_a=*/false, a, /*neg_b=*/false, b,
        /*c_mod=*/(short)0, acc, /*reuse_a=*/false, /*reuse_b=*/false);
  }
  const float bn = bias[n];
#pragma unroll
  for (int j = 0; j < 8; ++j) {
    int mo = rowBase + (half << 3) + j;
    Y[mo * HC + n] = acc[j] + bn;
  }
}

// ---------------------------------------------------------------------------
// K4: initialize out = bias (broadcast) and denom = 0.
// ---------------------------------------------------------------------------
__global__ void init_out_denom(float* __restrict__ out,
                               const float* __restrict__ bias,
                               float* __restrict__ denom) {
  int i = blockIdx.x * blockDim.x + threadIdx.x;
  if (i < N_NODES * HC) out[i] = bias[i & (HC - 1)];
  if (i < N_NODES * 2)  denom[i] = 0.f;
}

// ---------------------------------------------------------------------------
// K5: per 16-edge tile (one wave): ee = edge_attr @ W_comb + b_comb via WMMA
// (K=16 -> 4 WMMAs per column tile), fuse m = ee + xl[src] + xr[dst],
// LeakyReLU, dot with att, cross-lane reduce, write exp(s) and atomicAdd
// the softmax denominator per (dst, head).
// ---------------------------------------------------------------------------
__global__ __launch_bounds__(256)
void edge_scores_wmma(const float* __restrict__ EA,      // [E,16]
                      const int*   __restrict__ ei,      // [2,E]
                      const float* __restrict__ W_comb,  // [16,128]
                      const float* __restrict__ b_comb,  // [128]
                      const float* __restrict__ xl,      // [N,128]
                      const float* __restrict__ xr,      // [N,128]
                      const float* __restrict__ att,     // [2,64]
                      float* __restrict__ sExp,          // [E,2]
                      float* __restrict__ denom) {       // [N,2]
  const int lane  = threadIdx.x;
  const int tile  = blockIdx.x * blockDim.y + threadIdx.y;  // 0..49999
  const int eBase = tile << 4;
  const int half  = lane >> 4;
  const int l15   = lane & 15;
  const int kgrp  = half << 1;
  const int mA    = eBase + l15;   // A-fragment edge row

  int sidx[8], didx[8];
#pragma unroll
  for (int j = 0; j < 8; ++j) {
    int r = eBase + (half << 3) + j;   // matches C/D row layout
    sidx[j] = ei[r];
    didx[j] = ei[N_EDGES + r];
  }

  float ph0[8], ph1[8];
#pragma unroll
  for (int j = 0; j < 8; ++j) { ph0[j] = 0.f; ph1[j] = 0.f; }

#pragma unroll
  for (int ct = 0; ct < 8; ++ct) {
    const int n = (ct << 4) + l15;
    v8f acc = {};
#pragma unroll
    for (int k0 = 0; k0 < EDIM; k0 += 4) {
      v2f a = *(const v2f*)(EA + mA * EDIM + k0 + kgrp);
      v2f b;
      b.x = W_comb[(k0 + kgrp)     * HC + n];
      b.y = W_comb[(k0 + kgrp + 1) * HC + n];
      acc = __builtin_amdgcn_wmma_f32_16x16x4_f32(
          false, a, false, b, (short)0, acc, false, false);
    }
    const float bc = b_comb[n];
    const float an = att[((ct >= 4) ? CH : 0) + (n & 63)];
#pragma unroll
    for (int j = 0; j < 8; ++j) {
      float v = acc[j] + bc + xl[sidx[j] * HC + n] + xr[didx[j] * HC + n];
      v = (v > 0.f) ? v : NEG_SLOPE * v;
      if (ct < 4) ph0[j] += an * v; else ph1[j] += an * v;
    }
  }

  // reduce each head's score across the 16 lanes of this half (wave32)
#pragma unroll
  for (int j = 0; j < 8; ++j) {
    float s0 = ph0[j], s1 = ph1[j];
#pragma unroll
    for (int msk = 1; msk <= 8; msk <<= 1) {
      s0 += __shfl_xor(s0, msk, 32);
      s1 += __shfl_xor(s1, msk, 32);
    }
    if (l15 == 0) {
      int e = eBase + (half << 3) + j;
      float e0 = __expf(s0), e1 = __expf(s1);
      sExp[e * 2 + 0] = e0;
      sExp[e * 2 + 1] = e1;
      int d = didx[j];
      atomicAdd(&denom[d * 2 + 0], e0);
      atomicAdd(&denom[d * 2 + 1], e1);
    }
  }
}

// ---------------------------------------------------------------------------
// K6: out[dst] += alpha * xl[src].  One wave per edge, 4 cols per lane.
// ---------------------------------------------------------------------------
__global__ __launch_bounds__(256)
void scatter_out(const int* __restrict__ ei,
                 const float* __restrict__ xl,
                 const float* __restrict__ sExp,
                 const float* __restrict__ denom,
                 float* __restrict__ out) {
  int gid = blockIdx.x * blockDim.x + threadIdx.x;
  int e = gid >> 5;
  if (e >= N_EDGES) return;
  int lane = gid & 31;
  int s = ei[e];
  int d = ei[N_EDGES + e];
  int h = lane >> 4;                       // cols 0..63 head0, 64..127 head1
  float alpha = sExp[e * 2 + h] / denom[d * 2 + h];
  int c0 = lane << 2;
  float4 v = *(const float4*)(xl + s * HC + c0);
  atomicAdd(&out[d * HC + c0 + 0], alpha * v.x);
  atomicAdd(&out[d * HC + c0 + 1], alpha * v.y);
  atomicAdd(&out[d * HC + c0 + 2], alpha * v.z);
  atomicAdd(&out[d * HC + c0 + 3], alpha * v.w);
}

// ---------------------------------------------------------------------------
extern "C" void kernel_launch(void* const* d_in, const int* in_sizes, int n_in,
                              void* d_out, int out_size, void* d_ws, size_t ws_size,
                              hipStream_t stream) {
  const float* x    = (const float*)d_in[0];
  const int*   ei   = (const int*)  d_in[1];
  const float* ea   = (const float*)d_in[2];
  const float* W_ep = (const float*)d_in[3];
  const float* b_ep = (const float*)d_in[4];
  const float* W_l  = (const float*)d_in[5];
  const float* b_l  = (const float*)d_in[6];
  const float* W_r  = (const float*)d_in[7];
  const float* b_r  = (const float*)d_in[8];
  const float* W_e  = (const float*)d_in[9];
  const float* att  = (const float*)d_in[10];
  const float* bias = (const float*)d_in[11];
  float* out = (float*)d_out;

  // workspace layout (floats): xl, xr, W_comb, b_comb, sExp, denom  (~58 MB)
  float* ws     = (float*)d_ws;
  float* xl     = ws;
  float* xr     = xl + (size_t)N_NODES * HC;
  float* W_comb = xr + (size_t)N_NODES * HC;
  float* b_comb = W_comb + EDIM * HC;
  float* sExp   = b_comb + HC;
  float* denom  = sExp + (size_t)N_EDGES * 2;

  combine_weights<<<1, 128, 0, stream>>>(W_ep, b_ep, W_e, W_comb, b_comb);

  dim3 blk(32, 8);
  gemm_bias_wmma<<<N_NODES / 16, blk, 0, stream>>>(x, W_l, b_l, xl, DIN);
  gemm_bias_wmma<<<N_NODES / 16, blk, 0, stream>>>(x, W_r, b_r, xr, DIN);

  init_out_denom<<<(N_NODES * HC + 255) / 256, 256, 0, stream>>>(out, bias, denom);

  edge_scores_wmma<<<N_EDGES / 16 / 8, blk, 0, stream>>>(
      ea, ei, W_comb, b_comb, xl, xr, att, sExp, denom);

  scatter_out<<<(N_EDGES * 32) / 256, 256, 0, stream>>>(ei, xl, sExp, denom, out);
}